// CRF_55327768708232
// MI455X (gfx1250) — compile-verified
//
#include <hip/hip_runtime.h>
#include <hip/hip_bf16.h>

#define SEQ   1024
#define BATCH 512
#define NT    48     // num tags
#define SCP   52     // padded LDS row stride (floats)

#define LOG2E 1.4426950408889634f
#define LN2   0.6931471805599453f

typedef __attribute__((ext_vector_type(2))) float v2f;
typedef __attribute__((ext_vector_type(8))) float v8f;

// ---------------------------------------------------------------------------
// Kernel 1: numerator (gold-path score) per batch element -> ws_num[b]
// ---------------------------------------------------------------------------
__global__ __launch_bounds__(256) void crf_numerator(
    const float* __restrict__ em, const int* __restrict__ tags,
    const int* __restrict__ mask, const float* __restrict__ startT,
    const float* __restrict__ endT, const float* __restrict__ trans,
    float* __restrict__ ws_num) {
  int b = blockIdx.x * blockDim.x + threadIdx.x;
  if (b >= BATCH) return;

  int prev = tags[b];                           // tags[0][b]
  float num = startT[prev] + em[(size_t)b * NT + prev];
  int msum = (mask[b] != 0) ? 1 : 0;

#pragma unroll 4
  for (int i = 1; i < SEQ; ++i) {
    int t  = tags[i * BATCH + b];
    int mk = mask[i * BATCH + b];
    float e  = em[((size_t)i * BATCH + b) * NT + t];
    float tr = trans[prev * NT + t];
    num += (tr + e) * (float)mk;
    msum += (mk != 0) ? 1 : 0;
    prev = t;
  }
  int last = tags[(msum - 1) * BATCH + b];
  ws_num[b] = num + endT[last];
}

// ---------------------------------------------------------------------------
// Kernel 2: forward algorithm. One block = 16 batch rows, 3 waves (one 16-col
// N-tile each). Per step: row-max -> p=exp(score-m) -> q = p @ exp(Tr) via
// V_WMMA_F32_16X16X4_F32 -> score = log(q)+m+em (masked).
// q is provably in [0.9, 53], so raw v_log_f32 / v_exp_f32 are safe (no
// denorm/range fix-up needed on the 1024-step critical path).
// ---------------------------------------------------------------------------
__global__ __launch_bounds__(96) void crf_forward(
    const float* __restrict__ em, const int* __restrict__ mask,
    const float* __restrict__ startT, const float* __restrict__ endT,
    const float* __restrict__ trans, const float* __restrict__ ws_num,
    float* __restrict__ ws_llh) {
  __shared__ float sc[16][SCP];     // running scores, 16 batch rows x 48 tags
  __shared__ float pb[16][SCP];     // p = exp(sc - m)
  __shared__ float mex[NT][NT];     // exp(transitions)
  __shared__ float pmax[3][16];
  __shared__ float psum[3][16];
  __shared__ float mrow[16];

  const int tid  = threadIdx.x;
  const int w    = tid >> 5;        // wave id 0..2 -> N-tile
  const int lane = tid & 31;
  const int hi   = lane >> 4;       // half-wave select
  const int l4   = lane & 15;
  const int bb   = blockIdx.x * 16; // batch base
  const int co   = w * 16;          // column offset of this wave's N-tile

  // Stage exp(transitions) and initial scores.
  for (int i = tid; i < NT * NT; i += 96)
    (&mex[0][0])[i] = __builtin_amdgcn_exp2f(trans[i] * LOG2E);
  for (int i = tid; i < 16 * NT; i += 96) {
    int r = i / NT, c = i % NT;
    sc[r][c] = startT[c] + em[(size_t)(bb + r) * NT + c];
  }
  __syncthreads();

  for (int t = 1; t < SEQ; ++t) {
    // phase 1: per-wave partial row max over its 16 columns
    if (lane < 16) {
      float m = sc[lane][co];
#pragma unroll
      for (int c = 1; c < 16; ++c) m = fmaxf(m, sc[lane][co + c]);
      pmax[w][lane] = m;
    }
    __syncthreads();
    // phase 2+3 fused: wave0 publishes mrow for the update phase; every lane
    // combines the 3 partials inline for its own exps (saves one barrier).
    if (w == 0 && lane < 16)
      mrow[lane] = fmaxf(fmaxf(pmax[0][lane], pmax[1][lane]), pmax[2][lane]);
#pragma unroll
    for (int i = 0; i < 8; ++i) {
      int e = i * 32 + lane;
      int r = e >> 4;
      int c = co + (e & 15);
      float m = fmaxf(fmaxf(pmax[0][r], pmax[1][r]), pmax[2][r]);
      pb[r][c] = __builtin_amdgcn_exp2f((sc[r][c] - m) * LOG2E);
    }
    __syncthreads();
    // phase 4: q(16x16) = p(16x48) @ mex(48x16-tile) via 12 chained WMMAs
    v8f acc = {};
#pragma unroll
    for (int ks = 0; ks < 12; ++ks) {
      int k0 = ks * 4 + hi * 2;               // A: lanes 0-15 K={0,1}, 16-31 K={2,3}
      v2f a, bf;
      a.x  = pb[l4][k0];
      a.y  = pb[l4][k0 + 1];
      bf.x = mex[k0][co + l4];                // B mirrors A's K split across halves
      bf.y = mex[k0 + 1][co + l4];
      acc = __builtin_amdgcn_wmma_f32_16x16x4_f32(
          false, a, false, bf, (short)0, acc, false, false);
    }
    // phase 5: score' = log(q) + m + em[t], masked update
#pragma unroll
    for (int v = 0; v < 8; ++v) {
      int r = v + hi * 8;                     // C/D layout: VGPR v -> row v / v+8
      int c = co + l4;
      int b = bb + r;
      float e  = em[((size_t)t * BATCH + b) * NT + c];
      int   mk = mask[t * BATCH + b];
      float old = sc[r][c];
      float ns  = __builtin_amdgcn_logf(acc[v]) * LN2 + mrow[r] + e;
      sc[r][c]  = (mk != 0) ? ns : old;
    }
    __syncthreads();
  }

  // Final: denominator = logsumexp(score + end), llh = num - den
  if (lane < 16) {
    float m = sc[lane][co] + endT[co];
#pragma unroll
    for (int c = 1; c < 16; ++c) m = fmaxf(m, sc[lane][co + c] + endT[co + c]);
    pmax[w][lane] = m;
  }
  __syncthreads();
  if (w == 0 && lane < 16)
    mrow[lane] = fmaxf(fmaxf(pmax[0][lane], pmax[1][lane]), pmax[2][lane]);
  __syncthreads();
  if (lane < 16) {
    float s = 0.f;
#pragma unroll
    for (int c = 0; c < 16; ++c)
      s += __builtin_amdgcn_exp2f((sc[lane][co + c] + endT[co + c] - mrow[lane]) * LOG2E);
    psum[w][lane] = s;
  }
  __syncthreads();
  if (w == 0 && lane < 16) {
    int b = bb + lane;
    float den = mrow[lane] +
                __builtin_amdgcn_logf(psum[0][lane] + psum[1][lane] + psum[2][lane]) * LN2;
    ws_llh[b] = ws_num[b] - den;
  }
}

// ---------------------------------------------------------------------------
// Kernel 3: deterministic mean over 512 llh values -> d_out[0]
// ---------------------------------------------------------------------------
__global__ __launch_bounds__(256) void crf_reduce(
    const float* __restrict__ ws_llh, float* __restrict__ out) {
  __shared__ float red[256];
  int tid = threadIdx.x;
  red[tid] = ws_llh[tid] + ws_llh[tid + 256];
  __syncthreads();
  for (int off = 128; off > 0; off >>= 1) {
    if (tid < off) red[tid] += red[tid + off];
    __syncthreads();
  }
  if (tid == 0) out[0] = red[0] * (1.0f / BATCH);
}

// ---------------------------------------------------------------------------
extern "C" void kernel_launch(void* const* d_in, const int* in_sizes, int n_in,
                              void* d_out, int out_size, void* d_ws, size_t ws_size,
                              hipStream_t stream) {
  const float* em     = (const float*)d_in[0];
  const int*   tags   = (const int*)d_in[1];
  const int*   mask   = (const int*)d_in[2];
  const float* startT = (const float*)d_in[3];
  const float* endT   = (const float*)d_in[4];
  const float* trans  = (const float*)d_in[5];

  float* ws_num = (float*)d_ws;
  float* ws_llh = ws_num + BATCH;

  crf_numerator<<<2, 256, 0, stream>>>(em, tags, mask, startT, endT, trans, ws_num);
  crf_forward<<<BATCH / 16, 96, 0, stream>>>(em, mask, startT, endT, trans,
                                             ws_num, ws_llh);
  crf_reduce<<<1, 256, 0, stream>>>(ws_llh, (float*)d_out);
}